// OutputLayerReverseSegments_42545946034651
// MI455X (gfx1250) — compile-verified
//
#include <hip/hip_runtime.h>

typedef __attribute__((ext_vector_type(2))) float v2f;
typedef __attribute__((ext_vector_type(8))) float v8f;

#define F_TOTAL 720
#define D_MODEL 256
#define T_TOTAL 63
#define XROW    (T_TOTAL * D_MODEL)          /* 16128 floats per (b,c) row */
#define WT_ELEMS (6 * D_MODEL * F_TOTAL)     /* gathered transposed weights */

static __device__ __forceinline__ int imin(int a, int b) { return a < b ? a : b; }

// token index within partition j for forecast position f
static __device__ __forceinline__ int part_token(int f, int j) {
  if (j == 5) return (f < 368) ? (f / 23) : (16 + (f - 368) / 22);  // 16x23 + 16x22
  return f / (720 >> j);
}
// row within W_j for forecast position f
static __device__ __forceinline__ int part_row(int f, int j) {
  if (j == 5) return (f < 368) ? (f % 23) : ((f - 368) % 22);
  return f % (720 >> j);
}

// Build WT[j][d][f] = W_j[m_j(f)][d]  (dense, f-contiguous -> coalesced B loads)
__global__ void prep_wt_kernel(const float* __restrict__ W0, const float* __restrict__ W1,
                               const float* __restrict__ W2, const float* __restrict__ W3,
                               const float* __restrict__ W4, const float* __restrict__ W5,
                               float* __restrict__ wt) {
  int idx = blockIdx.x * blockDim.x + threadIdx.x;
  if (idx >= WT_ELEMS) return;
  int j   = idx / (D_MODEL * F_TOTAL);
  int rem = idx - j * (D_MODEL * F_TOTAL);
  int d   = rem / F_TOTAL;
  int f   = rem - d * F_TOTAL;
  int m   = part_row(f, j);
  const float* W = (j == 0) ? W0 : (j == 1) ? W1 : (j == 2) ? W2
                 : (j == 3) ? W3 : (j == 4) ? W4 : W5;
  wt[idx] = W[m * D_MODEL + d];
}

__global__ void prep_bias_kernel(const float* __restrict__ b0, const float* __restrict__ b1,
                                 const float* __restrict__ b2, const float* __restrict__ b3,
                                 const float* __restrict__ b4, const float* __restrict__ b5,
                                 float* __restrict__ bias) {
  int f = blockIdx.x * blockDim.x + threadIdx.x;
  if (f >= F_TOTAL) return;
  bias[f] = b0[part_row(f, 0)] + b1[part_row(f, 1)] + b2[part_row(f, 2)]
          + b3[part_row(f, 3)] + b4[part_row(f, 4)] + b5[part_row(f, 5)];
}

// Workgroup: 256 threads = 8 waves, output tile 128(bc) x 32(f).
// Wave (wm,wn): wm in 0..3 -> 32 M-rows, wn in 0..1 -> 16 N-cols; two 16x16 acc tiles.
__global__ __launch_bounds__(256) void proj_wmma_kernel(
    const float* __restrict__ x, const float* __restrict__ wt,
    const float* __restrict__ bias, float* __restrict__ out) {
  __shared__ __align__(16) float As[128 * 36];  // x chunk, pitch 36 (conflict-free, b64-aligned)
  __shared__ __align__(16) float Bs[32 * 34];   // W chunk transposed [n][k], pitch 34
  __shared__ int tokcol[32];                    // per-column token id for current partition

  const int tid  = threadIdx.x;
  const int lane = tid & 31;
  const int wave = tid >> 5;
  const int wm   = wave & 3;
  const int wn   = wave >> 2;
  const int bc0  = blockIdx.y * 128;
  const int f0   = blockIdx.x * 32;

  const int frag_mn = lane & 15;          // A row / B col for this lane
  const int frag_kh = (lane >> 4) * 2;    // lane-half K base within K=4 step

  v8f acc0 = {};
  v8f acc1 = {};

  // staging coordinates
  const int ar  = tid >> 3;               // 0..31 row group for A
  const int ak4 = (tid & 7) * 4;          // float4 column for A
  const int bk  = tid >> 3;               // 0..31 k for B
  const int bn4 = (tid & 7) * 4;          // 0..28 n base for B

  for (int j = 0; j < 6; ++j) {
    __syncthreads();
    if (tid < 32) tokcol[tid] = part_token(imin(f0 + tid, F_TOTAL - 1), j);
    __syncthreads();
    const int t0  = tokcol[0];
    const int t1  = tokcol[31];
    const int wt0 = tokcol[wn * 16];          // wave-uniform: this wave's column token range
    const int wt1 = tokcol[wn * 16 + 15];

    for (int t = t0; t <= t1; ++t) {          // one pass per token overlapping the tile
      const int gt = ((1 << j) - 1) + t;      // global token index (cumsum of k_j)
      const float* xa  = x  + (size_t)bc0 * XROW + (size_t)gt * D_MODEL;
      const float* wtp = wt + (size_t)j * (D_MODEL * F_TOTAL);
      const bool live = (t >= wt0) && (t <= wt1);  // wave-uniform -> EXEC stays all-1s

      for (int kk = 0; kk < D_MODEL; kk += 32) {
        __syncthreads();  // previous iteration's LDS reads complete
        // ---- stage A: x rows bc0..bc0+127, cols kk..kk+31 (coalesced b128) ----
        #pragma unroll
        for (int rr = 0; rr < 4; ++rr) {
          const int r = ar + 32 * rr;
          const float4 v = *(const float4*)(xa + (size_t)r * XROW + (kk + ak4));
          *(float4*)(&As[r * 36 + ak4]) = v;
        }
        // ---- stage B transposed + masked to current token ----
        #pragma unroll
        for (int e = 0; e < 4; ++e) {
          const int n = bn4 + e;
          const int f = f0 + n;
          float val = 0.0f;
          if (f < F_TOTAL && tokcol[n] == t)
            val = wtp[(kk + bk) * F_TOTAL + f];
          Bs[n * 34 + bk] = val;
        }
        __syncthreads();
        if (live) {
          const float* A0 = &As[(wm * 32 + frag_mn) * 36];
          const float* A1 = A0 + 16 * 36;
          const float* Bb = &Bs[(wn * 16 + frag_mn) * 34];
          #pragma unroll
          for (int k4 = 0; k4 < 32; k4 += 4) {
            const v2f a0 = *(const v2f*)(A0 + k4 + frag_kh);
            const v2f a1 = *(const v2f*)(A1 + k4 + frag_kh);
            const v2f bf = *(const v2f*)(Bb + k4 + frag_kh);
            acc0 = __builtin_amdgcn_wmma_f32_16x16x4_f32(
                false, a0, false, bf, (short)0, acc0, false, false);
            acc1 = __builtin_amdgcn_wmma_f32_16x16x4_f32(
                false, a1, false, bf, (short)0, acc1, false, false);
          }
        }
      }
    }
  }

  // ---- transpose through LDS for c-contiguous (coalesced) stores, fuse bias ----
  __syncthreads();
  #pragma unroll
  for (int r = 0; r < 8; ++r) {
    const int m = r + (lane >> 4) * 8;   // C/D layout: VGPR r holds rows r and r+8
    As[(wm * 32 + m)      * 36 + (wn * 16 + frag_mn)] = acc0[r];
    As[(wm * 32 + 16 + m) * 36 + (wn * 16 + frag_mn)] = acc1[r];
  }
  __syncthreads();
  const int rloc = tid & 127;
  const int row  = bc0 + rloc;
  const int bidx = row >> 6;
  const int cidx = row & 63;
  const int nb   = (tid >> 7) * 16;
  #pragma unroll
  for (int nn = 0; nn < 16; ++nn) {
    const int f = f0 + nb + nn;
    if (f < F_TOTAL)
      out[((size_t)bidx * F_TOTAL + f) * 64 + cidx] = As[rloc * 36 + nb + nn] + bias[f];
  }
}

extern "C" void kernel_launch(void* const* d_in, const int* in_sizes, int n_in,
                              void* d_out, int out_size, void* d_ws, size_t ws_size,
                              hipStream_t stream) {
  const float* x = (const float*)d_in[0];
  const float* W[6];
  const float* bb[6];
  for (int j = 0; j < 6; ++j) {
    W[j]  = (const float*)d_in[1 + 2 * j];
    bb[j] = (const float*)d_in[2 + 2 * j];
  }
  float* wt   = (float*)d_ws;          // 6*256*720 floats = 4.42 MB
  float* bias = wt + WT_ELEMS;         // +720 floats

  prep_wt_kernel<<<WT_ELEMS / 256, 256, 0, stream>>>(W[0], W[1], W[2], W[3], W[4], W[5], wt);
  prep_bias_kernel<<<3, 256, 0, stream>>>(bb[0], bb[1], bb[2], bb[3], bb[4], bb[5], bias);

  dim3 grid((F_TOTAL + 31) / 32, 4096 / 128);  // 23 x 32 workgroups
  proj_wmma_kernel<<<grid, 256, 0, stream>>>(x, wt, bias, (float*)d_out);
}